// GMM_4810363372989
// MI455X (gfx1250) — compile-verified
//
#include <hip/hip_runtime.h>
#include <hip/hip_bf16.h>

// GMM EM on gfx1250, fully fused per-iteration pass using fp32 WMMA
// (V_WMMA_F32_16X16X4_F32). B=8, N=80000, K=5, F=20, 5 EM iters + final E.
//
// Register-layout assumptions (from CDNA5 ISA, wave32):
//   A 16x4 f32 : lane (L=lane&15, h=lane>>4): v0=A[L][2h], v1=A[L][2h+1]
//   B 4x16 f32 : lane (j=lane&15, h):         v0=B[2h][j], v1=B[2h+1][j]
//   C/D 16x16  : lane (j,h), vgpr v:          D[v+8h][j]

typedef __attribute__((ext_vector_type(2))) float v2f;
typedef __attribute__((ext_vector_type(8))) float v8f;

static __device__ __forceinline__ v8f wmma4(v2f a, v2f b, v8f c) {
  return __builtin_amdgcn_wmma_f32_16x16x4_f32(false, a, false, b, (short)0, c,
                                               false, false);
}

#define GMM_B 8
#define GMM_N 80000
#define GMM_K 5
#define GMM_F 20
#define BLKS_PER_B 50          // 50 blocks per batch
#define TILES_PER_BLK 100      // 100 tiles of 16 rows -> 50*100*16 = 80000
#define OUT_LL   0
#define OUT_POST 3200000
#define OUT_MEAN 6400000
#define OUT_VAR  6400800
#define OUT_PI   6401600

// ---------------------------------------------------------------------------
// Prep: per (b,k) E-step coefficients.
//   cf[b][k][f<20]    = mu*alpha         (coeff of x)
//   cf[b][k][20+f]    = -0.5*alpha       (coeff of x^2)
//   cst[b][k]         = log(pi) - 0.5*sum(log(2*pi*var)) - 0.5*sum(mu^2*alpha)
// alpha = 1/(var + 1e-6). Rows k>=5 zeroed.
// ---------------------------------------------------------------------------
__global__ void gmm_prep(const float* __restrict__ mean,
                         const float* __restrict__ var,
                         const float* __restrict__ pi,
                         float* __restrict__ cf, float* __restrict__ cst) {
  const int b = blockIdx.x;
  const int t = threadIdx.x;  // 128 threads
  const float* mb = mean + b * GMM_K * GMM_F;
  const float* vb = var + b * GMM_K * GMM_F;
  for (int idx = t; idx < 16 * 40; idx += 128) {
    const int k = idx / 40, f2 = idx % 40;
    float val = 0.0f;
    if (k < GMM_K) {
      const int f = (f2 < 20) ? f2 : (f2 - 20);
      const float a = 1.0f / (vb[k * GMM_F + f] + 1e-6f);
      val = (f2 < 20) ? mb[k * GMM_F + f] * a : (-0.5f * a);
    }
    cf[(size_t)b * 640 + idx] = val;
  }
  if (t < 16) {
    float c = 0.0f;
    if (t < GMM_K) {
      float slog = 0.0f, q = 0.0f;
      for (int f = 0; f < GMM_F; ++f) {
        const float v = vb[t * GMM_F + f];
        const float a = 1.0f / (v + 1e-6f);
        const float mu = mb[t * GMM_F + f];
        slog += logf(2.0f * 3.14159265358979323846f * v);
        q += mu * mu * a;
      }
      c = logf(pi[b * GMM_K + t]) - 0.5f * slog - 0.5f * q;
    }
    cst[b * 16 + t] = c;
  }
}

// ---------------------------------------------------------------------------
// Fused E-step (+ M-step partial sums) over data. One block = 100 tiles of 16
// rows; 8 waves, each wave keeps three WMMA f32 accumulators:
//   s0 -> S1[f'=0..15], s1 -> S1[16..19]|S2[0..11], s2 -> S2[12..19]|S0(col 8)
// On finalIter, writes ll/post to d_out instead of accumulating.
// ---------------------------------------------------------------------------
__global__ __launch_bounds__(256) void gmm_estep(
    const float* __restrict__ data, const float* __restrict__ cf,
    const float* __restrict__ cst, float* __restrict__ part,
    float* __restrict__ out_ll, float* __restrict__ out_post, int finalIter) {
  __shared__ __align__(16) float xlds[8][16 * GMM_F];  // per-wave x tile
  __shared__ __align__(16) float plds[8][16][16];      // per-wave post tile
  const int tid = threadIdx.x;
  const int w = tid >> 5;
  const int lane = tid & 31;
  const int j = lane & 15;   // col index / row-of-tile for A loads
  const int h = lane >> 4;   // lane half
  const int blk = blockIdx.x;
  const int b = blk / BLKS_PER_B;
  const int bb = blk % BLKS_PER_B;

  // Hoist B operands (tile-invariant): bv[c]   = cf[j][4c+2h .. +1]   (x)
  //                                    bv[5+c] = cf[j][20+4c+2h .. +1] (x^2)
  const v2f* cfr = (const v2f*)(cf + ((size_t)b * 16 + j) * 40);
  v2f bv[10];
#pragma unroll
  for (int c = 0; c < 10; ++c) bv[c] = cfr[2 * c + h];
  const float cj = cst[b * 16 + j];

  v8f s0 = {0, 0, 0, 0, 0, 0, 0, 0};
  v8f s1 = {0, 0, 0, 0, 0, 0, 0, 0};
  v8f s2 = {0, 0, 0, 0, 0, 0, 0, 0};
  const float* dbase = data + (size_t)b * GMM_N * GMM_F;
  float* xl = xlds[w];

  for (int ti = w; ti < TILES_PER_BLK; ti += 8) {
    const int n0 = (bb * TILES_PER_BLK + ti) * 16;
    // A loads: lane (j,h) grabs x[row=j][4c+2h], x[row=j][4c+2h+1]
    const v2f* dp = (const v2f*)(dbase + (size_t)(n0 + j) * GMM_F);
    v2f xv[5];
#pragma unroll
    for (int c = 0; c < 5; ++c) xv[c] = dp[2 * c + h];

    // ---- E-step GEMM: 10 x V_WMMA_F32_16X16X4_F32 ----
    v8f d = {0, 0, 0, 0, 0, 0, 0, 0};
#pragma unroll
    for (int c = 0; c < 5; ++c) d = wmma4(xv[c], bv[c], d);
#pragma unroll
    for (int c = 0; c < 5; ++c) {
      v2f a2;
      a2.x = xv[c].x * xv[c].x;
      a2.y = xv[c].y * xv[c].y;
      d = wmma4(a2, bv[5 + c], d);
    }

    // ---- softmax over K=5 clusters (lanes base..base+4 of each half) ----
    const int base = h << 4;
    float pv[8], lv[8];
#pragma unroll
    for (int v = 0; v < 8; ++v) {
      const float x = d[v] + cj;
      float m = __shfl(x, base + 0);
      m = fmaxf(m, __shfl(x, base + 1));
      m = fmaxf(m, __shfl(x, base + 2));
      m = fmaxf(m, __shfl(x, base + 3));
      m = fmaxf(m, __shfl(x, base + 4));
      const float e = __expf(x - m);
      float s = __shfl(e, base + 0);
      s += __shfl(e, base + 1);
      s += __shfl(e, base + 2);
      s += __shfl(e, base + 3);
      s += __shfl(e, base + 4);
      pv[v] = (j < GMM_K) ? (e / s) : 0.0f;  // zero junk cluster rows
      lv[v] = x;
    }

    if (finalIter) {
      if (j < GMM_K) {
        const size_t o = ((size_t)b * GMM_N + (size_t)n0) * GMM_K + (size_t)j;
#pragma unroll
        for (int v = 0; v < 8; ++v) {
          const size_t oo = o + (size_t)(v + 8 * h) * GMM_K;
          out_ll[oo] = lv[v];
          out_post[oo] = pv[v];
        }
      }
    } else {
      // Stage tiles in wave-private LDS, then M-step GEMM (12 WMMAs).
#pragma unroll
      for (int c = 0; c < 5; ++c)
        *(v2f*)&xl[j * GMM_F + 4 * c + 2 * h] = xv[c];
#pragma unroll
      for (int v = 0; v < 8; ++v) plds[w][j][v + 8 * h] = pv[v];
      asm volatile("s_wait_dscnt 0" ::: "memory");  // wave-local LDS sync
#pragma unroll
      for (int cp = 0; cp < 4; ++cp) {
        const int r0 = 4 * cp + 2 * h, r1 = r0 + 1;
        v2f a;  // A = post^T: A[cluster=j][inner=row]
        a.x = plds[w][j][r0];
        a.y = plds[w][j][r1];
        {  // col-tile 0: f' = j  -> S1[f 0..15]
          v2f f0;
          f0.x = xl[r0 * GMM_F + j];
          f0.y = xl[r1 * GMM_F + j];
          s0 = wmma4(a, f0, s0);
        }
        {  // col-tile 1: f' = 16+j -> S1[16..19] | S2[0..11]
          const int fi = (j < 4) ? (16 + j) : (j - 4);
          float u0 = xl[r0 * GMM_F + fi], u1 = xl[r1 * GMM_F + fi];
          u0 = (j < 4) ? u0 : u0 * u0;
          u1 = (j < 4) ? u1 : u1 * u1;
          v2f f1;
          f1.x = u0;
          f1.y = u1;
          s1 = wmma4(a, f1, s1);
        }
        {  // col-tile 2: f' = 32+j -> S2[12..19] | S0 (j==8) | 0
          const int fi = (j < 8) ? (12 + j) : 0;
          float u0 = xl[r0 * GMM_F + fi], u1 = xl[r1 * GMM_F + fi];
          u0 *= u0;
          u1 *= u1;
          u0 = (j == 8) ? 1.0f : ((j > 8) ? 0.0f : u0);
          u1 = (j == 8) ? 1.0f : ((j > 8) ? 0.0f : u1);
          v2f f2;
          f2.x = u0;
          f2.y = u1;
          s2 = wmma4(a, f2, s2);
        }
      }
    }
  }

  if (!finalIter) {
    // Deterministic block reduction: per-wave partials -> LDS -> fixed-order
    // sum over the 8 waves -> per-block slot in global workspace.
    __syncthreads();  // done using xlds as x-tiles
    if (h == 0) {
      float* pl = xlds[w];  // reuse as [5][48]
#pragma unroll
      for (int v = 0; v < GMM_K; ++v) {
        pl[v * 48 + j] = s0[v];
        pl[v * 48 + 16 + j] = s1[v];
        pl[v * 48 + 32 + j] = s2[v];
      }
    }
    __syncthreads();
    for (int idx = tid; idx < 240; idx += 256) {
      float s = 0.0f;
#pragma unroll
      for (int ww = 0; ww < 8; ++ww) s += xlds[ww][idx];
      part[(size_t)blk * 240 + idx] = s;
    }
  }
}

// ---------------------------------------------------------------------------
// M-step finalize: fixed-order sum of the 50 block partials per batch, then
//   pi = L1-normalized sizes/N, mu = S1/(S0+1e-7),
//   var = (S2 - 2*mu*S1 + mu^2*S0)/(S0+1e-7) + 1e-6
// ---------------------------------------------------------------------------
__global__ __launch_bounds__(256) void gmm_mfin(
    const float* __restrict__ part, float* __restrict__ mean,
    float* __restrict__ var, float* __restrict__ pi, float* __restrict__ out,
    int writeOut) {
  __shared__ float S[240];  // [k][48]: f'<20 = S1, 20..39 = S2, 40 = S0
  const int b = blockIdx.x;
  const int t = threadIdx.x;
  const float* pb = part + (size_t)b * BLKS_PER_B * 240;
  for (int idx = t; idx < 240; idx += 256) {
    float s = 0.0f;
    for (int p = 0; p < BLKS_PER_B; ++p) s += pb[(size_t)p * 240 + idx];
    S[idx] = s;
  }
  __syncthreads();
  if (t == 0) {
    float pt[GMM_K];
    float l1 = 0.0f;
    for (int k = 0; k < GMM_K; ++k) {
      pt[k] = S[k * 48 + 40] / (float)GMM_N;
      l1 += fabsf(pt[k]);
    }
    l1 = fmaxf(l1, 1e-12f);
    for (int k = 0; k < GMM_K; ++k) {
      const float p = pt[k] / l1;
      pi[b * GMM_K + k] = p;
      if (writeOut) out[OUT_PI + b * GMM_K + k] = p;
    }
  }
  if (t < GMM_K * GMM_F) {
    const int k = t / GMM_F, f = t % GMM_F;
    const float sk = S[k * 48 + 40];
    const float a1 = S[k * 48 + f];
    const float a2 = S[k * 48 + 20 + f];
    const float denom = sk + 1e-7f;
    const float mu = a1 / denom;
    const float vv = (a2 - 2.0f * mu * a1 + mu * mu * sk) / denom + 1e-6f;
    mean[b * 100 + t] = mu;
    var[b * 100 + t] = vv;
    if (writeOut) {
      out[OUT_MEAN + b * 100 + t] = mu;
      out[OUT_VAR + b * 100 + t] = vv;
    }
  }
}

extern "C" void kernel_launch(void* const* d_in, const int* in_sizes, int n_in,
                              void* d_out, int out_size, void* d_ws,
                              size_t ws_size, hipStream_t stream) {
  const float* data = (const float*)d_in[0];
  const float* means0 = (const float*)d_in[1];
  const float* var0 = (const float*)d_in[2];
  const float* pi0 = (const float*)d_in[3];
  float* out = (float*)d_out;

  // Workspace layout (floats): total ~103k floats (~412 KB)
  float* ws = (float*)d_ws;
  float* wcf = ws;                   // 8*16*40 = 5120
  float* wcst = wcf + 5120;          // 8*16   = 128
  float* wmean = wcst + 128;         // 8*5*20 = 800
  float* wvar = wmean + 800;         // 800
  float* wpi = wvar + 800;           // 40
  float* wpart = wpi + 40;           // 400*240 = 96000

  float* out_ll = out + OUT_LL;
  float* out_post = out + OUT_POST;

  for (int it = 0; it <= 5; ++it) {
    const float* m = (it == 0) ? means0 : wmean;
    const float* v = (it == 0) ? var0 : wvar;
    const float* p = (it == 0) ? pi0 : wpi;
    gmm_prep<<<GMM_B, 128, 0, stream>>>(m, v, p, wcf, wcst);
    gmm_estep<<<GMM_B * BLKS_PER_B, 256, 0, stream>>>(
        data, wcf, wcst, wpart, out_ll, out_post, (it == 5) ? 1 : 0);
    if (it < 5)
      gmm_mfin<<<GMM_B, 256, 0, stream>>>(wpart, wmean, wvar, wpi, out,
                                          (it == 4) ? 1 : 0);
  }
}